// CrossEncoderMTL_29137058136523
// MI455X (gfx1250) — compile-verified
//
#include <hip/hip_runtime.h>
#include <hip/hip_bf16.h>

#define L_    4
#define B_    256
#define S_    256
#define H_    1024
#define K_    64
#define NSEC_ 26
#define SSPLIT 8

typedef __attribute__((ext_vector_type(2))) float v2f;
typedef __attribute__((ext_vector_type(4))) float v4f;
typedef __attribute__((ext_vector_type(8))) float v8f;

// ---------------------------------------------------------------------------
// Phase 1a: weighted layer pool + masked partial-sum over an S chunk.
// grid (B, SSPLIT), block 256. Thread t owns h = 4t..4t+3 (float4 lanes).
// hidden_states read exactly once -> non-temporal loads.
// ---------------------------------------------------------------------------
__global__ void pool_partial_kernel(const float* __restrict__ hs,
                                    const int*   __restrict__ mask,
                                    const float* __restrict__ lw,
                                    float*       __restrict__ partial) {
    const int b  = blockIdx.x;
    const int sp = blockIdx.y;
    const int t  = threadIdx.x;
    const int h  = t * 4;

    // softmax over the 4 layer weights (uniform scalar work)
    float w0 = lw[0], w1 = lw[1], w2 = lw[2], w3 = lw[3];
    float mx = fmaxf(fmaxf(w0, w1), fmaxf(w2, w3));
    float e0 = __expf(w0 - mx), e1 = __expf(w1 - mx);
    float e2 = __expf(w2 - mx), e3 = __expf(w3 - mx);
    float inv = 1.0f / (e0 + e1 + e2 + e3);
    float w[L_] = { e0 * inv, e1 * inv, e2 * inv, e3 * inv };

    v4f acc = {0.f, 0.f, 0.f, 0.f};
    const int s0 = sp * (S_ / SSPLIT);
    for (int s = s0; s < s0 + S_ / SSPLIT; ++s) {
        if (mask[b * S_ + s] != 0) {   // uniform branch per (b,s)
#pragma unroll
            for (int l = 0; l < L_; ++l) {
                const v4f* p = (const v4f*)(hs +
                    (((size_t)(l * B_ + b) * S_ + s) * H_ + h));
                v4f v = __builtin_nontemporal_load(p);
                acc += w[l] * v;
            }
        }
    }
    *(v4f*)(partial + ((size_t)(b * SSPLIT + sp) * H_ + h)) = acc;
}

// ---------------------------------------------------------------------------
// Phase 1b: reduce SSPLIT partials, divide by clipped mask count -> feats[B,H]
// grid B, block 256.
// ---------------------------------------------------------------------------
__global__ void finalize_feats_kernel(const float* __restrict__ partial,
                                      const int*   __restrict__ mask,
                                      float*       __restrict__ feats) {
    __shared__ int red[256];
    const int b = blockIdx.x;
    const int t = threadIdx.x;

    red[t] = mask[b * S_ + t];
    __syncthreads();
    for (int o = 128; o > 0; o >>= 1) {
        if (t < o) red[t] += red[t + o];
        __syncthreads();
    }
    const float rdenom = 1.0f / fmaxf((float)red[0], 1e-6f);

    const int h = t * 4;
    v4f acc = {0.f, 0.f, 0.f, 0.f};
#pragma unroll
    for (int sp = 0; sp < SSPLIT; ++sp)
        acc += *(const v4f*)(partial + ((size_t)(b * SSPLIT + sp) * H_ + h));
    acc *= rdenom;
    *(v4f*)(feats + (size_t)b * H_ + h) = acc;
}

// ---------------------------------------------------------------------------
// Phase 2a: down-proj + exact GELU, section-routed via masked commit.
// grid (NSEC, B/16), block 128 (4 waves; wave w owns N-tile w of K_=64).
// Per wave: D(16x16) = feats(16x1024) x Wd[sec](1024x16), K-loop of 256
// V_WMMA_F32_16X16X4_F32. Rows committed only where section_id matches.
// ---------------------------------------------------------------------------
__global__ void down_proj_kernel(const float* __restrict__ feats,
                                 const float* __restrict__ Wd,
                                 const float* __restrict__ bd,
                                 const int*   __restrict__ sec_id,
                                 float*       __restrict__ hbuf) {
    const int sec   = blockIdx.x;
    const int mtile = blockIdx.y;
    const int wave  = threadIdx.x >> 5;
    const int lane  = threadIdx.x & 31;
    const int ln    = lane & 15;      // column (B/C/D) or row (A)
    const int lh    = lane >> 4;      // lane-half selects K-pair / M-half
    const int n     = wave * 16 + ln; // 0..63
    const int b0    = mtile * 16;

    const float bias = bd[sec * K_ + n];
    v8f c;
#pragma unroll
    for (int v = 0; v < 8; ++v) c[v] = bias;

    const float* arow = feats + (size_t)(b0 + ln) * H_;
    const float* bcol = Wd + (size_t)sec * H_ * K_ + n;
    for (int k0 = 0; k0 < H_; k0 += 4) {
        const int ka = k0 + 2 * lh;
        v2f a = *(const v2f*)(arow + ka);          // A[m=ln][ka], A[m][ka+1]
        v2f bf;
        bf[0] = bcol[(size_t)ka * K_];             // B[ka][n]
        bf[1] = bcol[(size_t)(ka + 1) * K_];       // B[ka+1][n]
        c = __builtin_amdgcn_wmma_f32_16x16x4_f32(
                false, a, false, bf, (short)0, c, false, false);
    }

#pragma unroll
    for (int v = 0; v < 8; ++v) {
        const int m = v + 8 * lh;
        const int b = b0 + m;
        const float x = c[v];
        const float g = 0.5f * x * (1.0f + erff(x * 0.70710678118654752f));
        if (sec_id[b] == sec) hbuf[b * K_ + n] = g;
    }
}

// ---------------------------------------------------------------------------
// Phase 2b: up-proj + residual + bias, section-routed via masked commit.
// grid (NSEC, B/16, 16), block 128 (wave w owns N-tile blockIdx.z*4+w of H).
// C initialized with residual feats + b_up; K-loop 64 -> 16 WMMAs.
// ---------------------------------------------------------------------------
__global__ void up_proj_kernel(const float* __restrict__ feats,
                               const float* __restrict__ hbuf,
                               const float* __restrict__ Wu,
                               const float* __restrict__ bu,
                               const int*   __restrict__ sec_id,
                               float*       __restrict__ feats2) {
    const int sec   = blockIdx.x;
    const int mtile = blockIdx.y;
    const int wave  = threadIdx.x >> 5;
    const int lane  = threadIdx.x & 31;
    const int ln    = lane & 15;
    const int lh    = lane >> 4;
    const int col   = (blockIdx.z * 4 + wave) * 16 + ln;   // 0..1023
    const int b0    = mtile * 16;

    const float bias = bu[sec * H_ + col];
    v8f c;
#pragma unroll
    for (int v = 0; v < 8; ++v) {
        const int m = v + 8 * lh;
        c[v] = feats[(size_t)(b0 + m) * H_ + col] + bias;  // residual + bias
    }

    const float* arow = hbuf + (size_t)(b0 + ln) * K_;
    const float* bcol = Wu + (size_t)sec * K_ * H_ + col;
    for (int k0 = 0; k0 < K_; k0 += 4) {
        const int ka = k0 + 2 * lh;
        v2f a = *(const v2f*)(arow + ka);
        v2f bf;
        bf[0] = bcol[(size_t)ka * H_];
        bf[1] = bcol[(size_t)(ka + 1) * H_];
        c = __builtin_amdgcn_wmma_f32_16x16x4_f32(
                false, a, false, bf, (short)0, c, false, false);
    }

#pragma unroll
    for (int v = 0; v < 8; ++v) {
        const int m = v + 8 * lh;
        const int b = b0 + m;
        if (sec_id[b] == sec) feats2[(size_t)b * H_ + col] = c[v];
    }
}

// ---------------------------------------------------------------------------
// Phase 3: heads. grid B, block 256. 5 dot products of length 1024 per sample.
// out[0:256] = reg, out[256 + b*4 + j] = ord_logits[b][j].
// ---------------------------------------------------------------------------
__global__ void heads_kernel(const float* __restrict__ feats2,
                             const float* __restrict__ Wreg,
                             const float* __restrict__ breg,
                             const float* __restrict__ Word,
                             const float* __restrict__ bord,
                             float*       __restrict__ out) {
    __shared__ float red[256];
    const int b = blockIdx.x;
    const int t = threadIdx.x;
    const int h = t * 4;

    v4f f = *(const v4f*)(feats2 + (size_t)b * H_ + h);
    float acc[5] = {0.f, 0.f, 0.f, 0.f, 0.f};
#pragma unroll
    for (int i = 0; i < 4; ++i) {
        const float x = f[i];
        const int hh = h + i;
        acc[0] += x * Wreg[hh];
        acc[1] += x * Word[hh * 4 + 0];
        acc[2] += x * Word[hh * 4 + 1];
        acc[3] += x * Word[hh * 4 + 2];
        acc[4] += x * Word[hh * 4 + 3];
    }
#pragma unroll
    for (int j = 0; j < 5; ++j) {
        red[t] = acc[j];
        __syncthreads();
        for (int o = 128; o > 0; o >>= 1) {
            if (t < o) red[t] += red[t + o];
            __syncthreads();
        }
        if (t == 0) {
            if (j == 0) out[b] = red[0] + breg[0];
            else        out[B_ + b * 4 + (j - 1)] = red[0] + bord[j - 1];
        }
        __syncthreads();
    }
}

// ---------------------------------------------------------------------------
extern "C" void kernel_launch(void* const* d_in, const int* in_sizes, int n_in,
                              void* d_out, int out_size, void* d_ws, size_t ws_size,
                              hipStream_t stream) {
    const float* hs    = (const float*)d_in[0];   // [L,B,S,H]
    const int*   mask  = (const int*)  d_in[1];   // [B,S]
    const int*   secid = (const int*)  d_in[2];   // [B]
    const float* lw    = (const float*)d_in[3];   // [L]
    const float* Wd    = (const float*)d_in[4];   // [NSEC,H,K]
    const float* bd    = (const float*)d_in[5];   // [NSEC,K]
    const float* Wu    = (const float*)d_in[6];   // [NSEC,K,H]
    const float* bu    = (const float*)d_in[7];   // [NSEC,H]
    const float* Wreg  = (const float*)d_in[8];   // [H,1]
    const float* breg  = (const float*)d_in[9];   // [1]
    const float* Word  = (const float*)d_in[10];  // [H,4]
    const float* bord  = (const float*)d_in[11];  // [4]
    float*       out   = (float*)d_out;           // [256 + 1024]

    float* wsf     = (float*)d_ws;
    float* partial = wsf;                                   // B*SSPLIT*H
    float* feats   = partial + (size_t)B_ * SSPLIT * H_;    // B*H
    float* hbuf    = feats   + (size_t)B_ * H_;             // B*K
    float* feats2  = hbuf    + (size_t)B_ * K_;             // B*H

    pool_partial_kernel<<<dim3(B_, SSPLIT), 256, 0, stream>>>(hs, mask, lw, partial);
    finalize_feats_kernel<<<B_, 256, 0, stream>>>(partial, mask, feats);
    down_proj_kernel<<<dim3(NSEC_, B_ / 16), 128, 0, stream>>>(feats, Wd, bd, secid, hbuf);
    up_proj_kernel<<<dim3(NSEC_, B_ / 16, H_ / 64), 128, 0, stream>>>(feats, hbuf, Wu, bu, secid, feats2);
    heads_kernel<<<B_, 256, 0, stream>>>(feats2, Wreg, breg, Word, bord, out);
}